// PointcloudGrouping_58909771432003
// MI455X (gfx1250) — compile-verified
//
#include <hip/hip_runtime.h>
#include <hip/hip_bf16.h>

// ---------------------------------------------------------------------------
// Problem constants (from reference)
// ---------------------------------------------------------------------------
#define BATCH        8
#define NPTS         16384
#define NCH          4
#define NUM_GROUPS   128
#define GROUP_SIZE   32
#define GROUP_RADIUS 0.1f
#define UPSCALE_K    128
#define OVERLAP_FAC  0.7f

#define CHUNK        512          // points per ball-query chunk
#define GT_GROUPS    16           // groups per ball-query block (one WMMA M-tile)

typedef __attribute__((ext_vector_type(2))) float v2f;
typedef __attribute__((ext_vector_type(8))) float v8f;

static __device__ __forceinline__ float neg_inf() { return -__builtin_inff(); }

// ---------------------------------------------------------------------------
// Kernel 1: Farthest point sampling. One block per batch, 1024 threads.
// Thread t owns points p = s*1024 + t, s in [0,16). mind kept in registers,
// coords re-read from global (L2 resident). Argmax tie-break: smallest index
// (matches jnp.argmax first-occurrence).
// ---------------------------------------------------------------------------
__global__ void __launch_bounds__(1024)
fps_kernel(const float* __restrict__ pts, const int* __restrict__ lengths,
           float* __restrict__ centers /* (B,128,3) */) {
    const int b = blockIdx.x;
    const int t = threadIdx.x;
    const int lane = t & 31;
    const int wv   = t >> 5;
    const int len  = lengths[b];
    const float4* __restrict__ P = (const float4*)(pts + (size_t)b * NPTS * NCH);

    float mind[16];
    #pragma unroll
    for (int s = 0; s < 16; ++s) {
        int p = s * 1024 + t;
        mind[s] = (p < len) ? __builtin_inff() : neg_inf();
    }

    __shared__ float sc[3];
    __shared__ float swv[32];
    __shared__ int   swi[32];
    __shared__ int   s_win;

    if (t == 0) {             // cur_0 = 0: first center is point 0
        float4 v = P[0];
        sc[0] = v.x; sc[1] = v.y; sc[2] = v.z;
    }
    __syncthreads();

    for (int k = 0; k < NUM_GROUPS; ++k) {
        const float cx = sc[0], cy = sc[1], cz = sc[2];
        if (t == 0) {
            float* cw = centers + (size_t)(b * NUM_GROUPS + k) * 3;
            cw[0] = cx; cw[1] = cy; cw[2] = cz;
        }
        float bestv = neg_inf();
        int   besti = 0x7fffffff;
        #pragma unroll
        for (int s = 0; s < 16; ++s) {
            int p = s * 1024 + t;
            float4 v = P[p];
            float dx = v.x - cx, dy = v.y - cy, dz = v.z - cz;
            float d = dx * dx + dy * dy + dz * dz;
            mind[s] = fminf(mind[s], d);
            if (mind[s] > bestv || (mind[s] == bestv && p < besti)) {
                bestv = mind[s]; besti = p;
            }
        }
        // wave32 reduce (value, min-index tiebreak)
        #pragma unroll
        for (int off = 16; off; off >>= 1) {
            float ov = __shfl_xor(bestv, off, 32);
            int   oi = __shfl_xor(besti, off, 32);
            if (ov > bestv || (ov == bestv && oi < besti)) { bestv = ov; besti = oi; }
        }
        if (lane == 0) { swv[wv] = bestv; swi[wv] = besti; }
        __syncthreads();
        if (wv == 0) {
            float v2 = swv[lane];
            int   i2 = swi[lane];
            #pragma unroll
            for (int off = 16; off; off >>= 1) {
                float ov = __shfl_xor(v2, off, 32);
                int   oi = __shfl_xor(i2, off, 32);
                if (ov > v2 || (ov == v2 && oi < i2)) { v2 = ov; i2 = oi; }
            }
            if (lane == 0) s_win = i2;
        }
        __syncthreads();
        const int win = s_win;
        if (t == (win & 1023)) {     // owner publishes next center
            float4 v = P[win];
            sc[0] = v.x; sc[1] = v.y; sc[2] = v.z;
        }
        __syncthreads();
    }
}

// ---------------------------------------------------------------------------
// Kernel 2: greedy cluster-NMS over FPS-ordered centers. One block per batch.
// ---------------------------------------------------------------------------
__global__ void __launch_bounds__(128)
cnms_kernel(const float* __restrict__ centers, int* __restrict__ retain) {
    const int b = blockIdx.x;
    const int t = threadIdx.x;
    __shared__ float cx[NUM_GROUPS], cy[NUM_GROUPS], cz[NUM_GROUPS];
    __shared__ int   keep[NUM_GROUPS];
    __shared__ int   flag;

    const float* c = centers + (size_t)(b * NUM_GROUPS + t) * 3;
    cx[t] = c[0]; cy[t] = c[1]; cz[t] = c[2];
    keep[t] = 0;
    if (t == 0) flag = 0;
    __syncthreads();

    float thr = 2.0f * GROUP_RADIUS * (1.0f - OVERLAP_FAC);
    thr *= thr;

    for (int j = 0; j < NUM_GROUPS; ++j) {
        float dx = cx[j] - cx[t], dy = cy[j] - cy[t], dz = cz[j] - cz[t];
        float d2 = dx * dx + dy * dy + dz * dz;
        if (keep[t] && d2 < thr) flag = 1;   // benign same-value race
        __syncthreads();
        if (t == 0) { keep[j] = flag ? 0 : 1; flag = 0; }
        __syncthreads();
    }
    retain[b * NUM_GROUPS + t] = keep[t];
}

// ---------------------------------------------------------------------------
// Kernel 3: ball query via V_WMMA_F32_16X16X4_F32.
//   d2 = |p|^2 - 2 c.p + |c|^2 :
//   A row m  = (-2cx, -2cy, -2cz, 1)          (16x4, f32)
//   B col n  = ( px,   py,   pz, |p|^2)       (4x16, f32)
//   C init   = |c_m|^2 broadcast per row
// Block = (batch, group-tile of 16) x 512 threads (16 waves).
// After each chunk's d2 tile lands in LDS, wave w stream-compacts the first
// 128 in-radius indices (in point order) for group gt*16+w.
// ---------------------------------------------------------------------------
__global__ void __launch_bounds__(512)
ballq_kernel(const float* __restrict__ pts, const int* __restrict__ lengths,
             const float* __restrict__ centers, const int* __restrict__ retain,
             int* __restrict__ ballidx /* (B,128,128) */) {
    const int b   = blockIdx.x;
    const int gt  = blockIdx.y;
    const int tid = threadIdx.x;
    const int lane = tid & 31;
    const int w    = tid >> 5;

    __shared__ float s_pt[CHUNK][4];              // x,y,z,|p|^2
    __shared__ float s_d2[GT_GROUPS * CHUNK];     // 32 KB
    __shared__ float s_ac[GT_GROUPS][4];          // A rows
    __shared__ float s_csq[GT_GROUPS];
    __shared__ int   s_ret[GT_GROUPS];

    const int len = lengths[b];
    if (tid < GT_GROUPS) {
        int g = gt * GT_GROUPS + tid;
        const float* c = centers + (size_t)(b * NUM_GROUPS + g) * 3;
        float x = c[0], y = c[1], z = c[2];
        s_ac[tid][0] = -2.0f * x;
        s_ac[tid][1] = -2.0f * y;
        s_ac[tid][2] = -2.0f * z;
        s_ac[tid][3] = 1.0f;
        s_csq[tid]   = x * x + y * y + z * z;
        s_ret[tid]   = retain[b * NUM_GROUPS + g];
    }
    // init this wave's output slots to -1 (wave w owns group gt*16+w)
    {
        int g = gt * GT_GROUPS + w;
        int* op = ballidx + (((size_t)b * NUM_GROUPS + g) << 7);
        #pragma unroll
        for (int j = 0; j < 4; ++j) op[j * 32 + lane] = -1;
    }
    __syncthreads();

    // Per-lane WMMA operand slots (ISA 7.12.2 layouts)
    const int mA  = lane & 15;
    const int kb  = (lane >> 4) * 2;
    const int hi8 = (lane >> 4) * 8;
    v2f a;
    a.x = s_ac[mA][kb];
    a.y = s_ac[mA][kb + 1];
    v8f cinit;
    #pragma unroll
    for (int r = 0; r < 8; ++r) cinit[r] = s_csq[r + hi8];

    const int g = gt * GT_GROUPS + w;
    int* outp = ballidx + (((size_t)b * NUM_GROUPS + g) << 7);
    const int ret = s_ret[w];
    int count = 0;
    const float R2 = GROUP_RADIUS * GROUP_RADIUS;
    const float4* __restrict__ P = (const float4*)(pts + (size_t)b * NPTS * NCH);

    for (int ch = 0; ch < NPTS / CHUNK; ++ch) {
        { // cooperative point-chunk load, precompute |p|^2
            float4 v = P[ch * CHUNK + tid];
            s_pt[tid][0] = v.x; s_pt[tid][1] = v.y; s_pt[tid][2] = v.z;
            s_pt[tid][3] = v.x * v.x + v.y * v.y + v.z * v.z;
        }
        __syncthreads();

        // 32 subtiles of 16 points; 2 per wave. EXEC is all-ones here.
        #pragma unroll
        for (int si = 0; si < 2; ++si) {
            const int sub = w * 2 + si;
            const int col = sub * 16 + mA;
            v2f bb;
            bb.x = s_pt[col][kb];
            bb.y = s_pt[col][kb + 1];
            v8f d = __builtin_amdgcn_wmma_f32_16x16x4_f32(
                false, a, false, bb, (short)0, cinit, false, false);
            #pragma unroll
            for (int r = 0; r < 8; ++r)
                s_d2[(r + hi8) * CHUNK + col] = d[r];
        }
        __syncthreads();

        // ordered first-K compaction for this wave's group
        for (int st = 0; st < CHUNK / 32; ++st) {
            int   p   = ch * CHUNK + st * 32 + lane;
            float d2v = s_d2[w * CHUNK + st * 32 + lane];
            bool hit = ret && (d2v <= R2) && (p < len);
            unsigned mask = (unsigned)__ballot(hit);
            int pos = count + __popc(mask & ((1u << lane) - 1u));
            if (hit && pos < UPSCALE_K) outp[pos] = p;
            count += __popc(mask);
        }
        __syncthreads();
    }
}

// ---------------------------------------------------------------------------
// Kernel 4: top-32 energies per group (stable, matches lax.top_k ties),
// gather + normalize rows exactly in reference order of operations.
// One wave per (b,g): block = 128 threads (4 waves), grid = (8, 32).
// ---------------------------------------------------------------------------
__global__ void __launch_bounds__(128)
topk_kernel(const float* __restrict__ pts, const float* __restrict__ centers,
            const int* __restrict__ ballidx, float* __restrict__ groups,
            int* __restrict__ flags) {
    const int b    = blockIdx.x;
    const int w    = threadIdx.x >> 5;
    const int lane = threadIdx.x & 31;
    const int g    = blockIdx.y * 4 + w;

    const int* ip = ballidx + (((size_t)b * NUM_GROUPS + g) << 7);
    const float* P = pts + (size_t)b * NPTS * NCH;

    int   idxl[4];
    float el[4];
    bool  taken[4];
    #pragma unroll
    for (int j = 0; j < 4; ++j) {
        int e  = lane * 4 + j;
        int id = ip[e];
        idxl[j]  = id;
        el[j]    = (id >= 0) ? P[(size_t)id * NCH + 3] : neg_inf();
        taken[j] = false;
    }
    const float* c = centers + (size_t)(b * NUM_GROUPS + g) * 3;
    const float cx = c[0], cy = c[1], cz = c[2];

    int allvalid = 1;
    int tidx0    = -1;
    for (int k = 0; k < GROUP_SIZE; ++k) {
        float bv = neg_inf();
        int   be = 0x7fffffff;
        #pragma unroll
        for (int j = 0; j < 4; ++j) {
            float v = taken[j] ? neg_inf()  : el[j];
            int   e = taken[j] ? 0x7fffffff : (lane * 4 + j);
            if (v > bv || (v == bv && e < be)) { bv = v; be = e; }
        }
        #pragma unroll
        for (int off = 16; off; off >>= 1) {
            float ov = __shfl_xor(bv, off, 32);
            int   oe = __shfl_xor(be, off, 32);
            if (ov > bv || (ov == bv && oe < be)) { bv = ov; be = oe; }
        }
        const int srcLane = be >> 2;
        const int sj      = be & 3;
        int cand = (sj == 0) ? idxl[0] : (sj == 1) ? idxl[1] : (sj == 2) ? idxl[2] : idxl[3];
        int tid_k = __shfl(cand, srcLane, 32);
        if (lane == srcLane) {
            if (sj == 0) taken[0] = true; else if (sj == 1) taken[1] = true;
            else if (sj == 2) taken[2] = true; else taken[3] = true;
        }
        const bool isneg = (bv == neg_inf());
        const int tidx   = isneg ? -1 : tid_k;
        if (k == 0) tidx0 = tidx;
        const int fidx = (tidx == -1) ? tidx0 : tidx;
        allvalid &= (tidx != -1) ? 1 : 0;

        if (lane == k) {
            float gx, gy, gz, ge;
            if (fidx < 0) { gx = gy = gz = ge = 0.0f; }
            else {
                const float* q = P + (size_t)fidx * NCH;
                gx = q[0]; gy = q[1]; gz = q[2]; ge = q[3];
            }
            // reference order: zero-mask, then subtract center (xyz), then /radius (all 4)
            gx = (gx - cx) / GROUP_RADIUS;
            gy = (gy - cy) / GROUP_RADIUS;
            gz = (gz - cz) / GROUP_RADIUS;
            ge = ge / GROUP_RADIUS;
            float* o = groups + ((((size_t)b * NUM_GROUPS + g) * GROUP_SIZE + k) << 2);
            o[0] = gx; o[1] = gy; o[2] = gz; o[3] = ge;
        }
    }
    if (lane == 0) flags[b * NUM_GROUPS + g] = allvalid;
}

// ---------------------------------------------------------------------------
// Kernel 5: embedding mask: arange(128) < (#groups with all 32 valid)
// ---------------------------------------------------------------------------
__global__ void __launch_bounds__(128)
mask_kernel(const int* __restrict__ flags, float* __restrict__ maskout) {
    const int b = blockIdx.x;
    const int t = threadIdx.x;
    __shared__ int s[NUM_GROUPS];
    s[t] = flags[b * NUM_GROUPS + t];
    __syncthreads();
    #pragma unroll
    for (int off = 64; off; off >>= 1) {
        if (t < off) s[t] += s[t + off];
        __syncthreads();
    }
    const int L = s[0];
    maskout[b * NUM_GROUPS + t] = (t < L) ? 1.0f : 0.0f;
}

// ---------------------------------------------------------------------------
// Launch
// d_out layout (floats): groups[8*128*32*4] | centers[8*128*3] | mask[8*128]
// d_ws  layout (ints)  : retain[1024] | ballidx[131072] | flags[1024]
// ---------------------------------------------------------------------------
extern "C" void kernel_launch(void* const* d_in, const int* in_sizes, int n_in,
                              void* d_out, int out_size, void* d_ws, size_t ws_size,
                              hipStream_t stream) {
    const float* pts     = (const float*)d_in[0];
    const int*   lengths = (const int*)d_in[1];

    float* out     = (float*)d_out;
    float* groups  = out;                                        // 131072
    float* centers = out + BATCH * NUM_GROUPS * GROUP_SIZE * NCH; // +3072
    float* maskout = centers + BATCH * NUM_GROUPS * 3;            // +1024

    int* retain  = (int*)d_ws;                                   // 1024
    int* ballidx = retain + BATCH * NUM_GROUPS;                  // 131072
    int* flags   = ballidx + BATCH * NUM_GROUPS * UPSCALE_K;     // 1024

    fps_kernel  <<<BATCH, 1024, 0, stream>>>(pts, lengths, centers);
    cnms_kernel <<<BATCH, 128,  0, stream>>>(centers, retain);
    ballq_kernel<<<dim3(BATCH, NUM_GROUPS / GT_GROUPS), 512, 0, stream>>>(
        pts, lengths, centers, retain, ballidx);
    topk_kernel <<<dim3(BATCH, NUM_GROUPS / 4), 128, 0, stream>>>(
        pts, centers, ballidx, groups, flags);
    mask_kernel <<<BATCH, 128, 0, stream>>>(flags, maskout);
}